// MBFCA_4183298146532
// MI455X (gfx1250) — compile-verified
//
#include <hip/hip_runtime.h>
#include <math.h>

typedef __attribute__((ext_vector_type(2))) float v2f;
typedef __attribute__((ext_vector_type(8))) float v8f;

namespace {
constexpr int kBands = 3;
constexpr int kB = 48;
constexpr int kS = 1024;
constexpr int kNPos = kBands * kB * kS;   // 147456 positions (band,b,s)
constexpr int kRows = kB * kS;            // 49152 rows of the (B*S, 9) matrix
// log2(e) / sqrt(3): softmax computed in base-2 domain (exactly equivalent,
// saves the per-exp v_mul by log2e the compiler would otherwise insert).
constexpr float kScale = 0.83298625576588513f;
}

// ---------------------------------------------------------------------------
// Kernel 1: per-band Q/K/V projections, padded d=3 -> 4 so WMMA fragments are
// plain aligned float2/float4 loads. Q is pre-scaled by log2(e)/sqrt(3) so the
// attention loop feeds raw WMMA output straight into v_exp_f32 (exp2).
// ---------------------------------------------------------------------------
__global__ __launch_bounds__(256) void mbfca_proj(
    const float* __restrict__ x,
    const float* __restrict__ WQ, const float* __restrict__ bQ,
    const float* __restrict__ WK, const float* __restrict__ bK,
    const float* __restrict__ WV, const float* __restrict__ bV,
    float* __restrict__ Qp, float* __restrict__ Kp, float* __restrict__ Vp)
{
  int idx = blockIdx.x * blockDim.x + threadIdx.x;
  if (idx >= kNPos) return;
  int n = idx / (kB * kS);                 // band
  float x0 = x[idx * 3 + 0];
  float x1 = x[idx * 3 + 1];
  float x2 = x[idx * 3 + 2];
  float q[3], k[3], v[3];
#pragma unroll
  for (int e = 0; e < 3; ++e) {
    const float* wq = WQ + (n * 3 + e) * 3;
    const float* wk = WK + (n * 3 + e) * 3;
    const float* wv = WV + (n * 3 + e) * 3;
    q[e] = fmaf(x0, wq[0], fmaf(x1, wq[1], fmaf(x2, wq[2], bQ[n * 3 + e]))) * kScale;
    k[e] = fmaf(x0, wk[0], fmaf(x1, wk[1], fmaf(x2, wk[2], bK[n * 3 + e])));
    v[e] = fmaf(x0, wv[0], fmaf(x1, wv[1], fmaf(x2, wv[2], bV[n * 3 + e])));
  }
  *(float4*)(Qp + (size_t)idx * 4) = make_float4(q[0], q[1], q[2], 0.0f);
  *(float4*)(Kp + (size_t)idx * 4) = make_float4(k[0], k[1], k[2], 0.0f);
  *(float4*)(Vp + (size_t)idx * 4) = make_float4(v[0], v[1], v[2], 0.0f);
}

// ---------------------------------------------------------------------------
// Kernel 2: flash attention. One wave32 per (attn n, batch b, 16-row Q tile).
// Scores via V_WMMA_F32_16X16X4_F32 (K=4 >= d=3, zero padded).
//
// Each lane owns the stride-16 column subset t = {t0+col} of the softmax axis,
// so the hot loop runs a LANE-LOCAL online softmax in base-2 (registers plus
// bare v_exp_f32 via __builtin_amdgcn_exp2f — no denorm-guard expansion, no
// cross-lane traffic). The 16 per-lane states per row are merged once in the
// epilogue with a 4-step butterfly (max, rescale, sum).
// ---------------------------------------------------------------------------
__global__ __launch_bounds__(256) void mbfca_attn(
    const float* __restrict__ Qp, const float* __restrict__ Kp,
    const float* __restrict__ Vp, float* __restrict__ O)
{
  const int wave = (blockIdx.x * 256 + (int)threadIdx.x) >> 5;
  const int lane = threadIdx.x & 31;
  const int half = lane >> 4;     // C-matrix half: rows M = r + 8*half
  const int col  = lane & 15;     // C-matrix column N (== t within tile)

  const int tile = wave & 63;                 // 64 tiles of 16 rows
  const int rem  = wave >> 6;
  const int b    = rem % kB;
  const int n    = rem / kB;                  // attention index 0..2
  const int qb = (n + 2) % 3;                 // a1:(Q2,K0,V1) a2:(Q0,K1,V2) a3:(Q1,K2,V0)
  const int kb = n;
  const int vb = (n + 1) % 3;

  const int s0 = tile * 16;

  // A fragment (Q tile 16x4): lane m = col, k in {2*half, 2*half+1}
  const float* qrow = Qp + (((size_t)qb * kB + b) * kS + (size_t)(s0 + col)) * 4 + 2 * half;
  v2f A = *(const v2f*)qrow;

  const float* kband = Kp + (((size_t)kb * kB + b) * kS) * 4;
  const float* vband = Vp + (((size_t)vb * kB + b) * kS) * 4;

  // lane-local online softmax state for the 8 rows this lane covers
  float m[8], l[8], a0[8], a1[8], a2[8];
#pragma unroll
  for (int r = 0; r < 8; ++r) { m[r] = -1e30f; l[r] = 0.f; a0[r] = 0.f; a1[r] = 0.f; a2[r] = 0.f; }

  for (int t0 = 0; t0 < kS; t0 += 16) {
    // B fragment (K^T tile 4x16): same per-lane pattern as A by layout symmetry
    v2f Bv = *(const v2f*)(kband + (size_t)(t0 + col) * 4 + 2 * half);
    float4 Vv = *(const float4*)(vband + (size_t)(t0 + col) * 4);

    v8f c = {0.f, 0.f, 0.f, 0.f, 0.f, 0.f, 0.f, 0.f};
    c = __builtin_amdgcn_wmma_f32_16x16x4_f32(false, A, false, Bv, (short)0, c, false, false);

#pragma unroll
    for (int r = 0; r < 8; ++r) {
      float sv   = c[r];                       // base-2 logit, row M=r+8*half
      float mnew = fmaxf(m[r], sv);
      float al   = __builtin_amdgcn_exp2f(m[r] - mnew);  // rescale of old state
      float p    = __builtin_amdgcn_exp2f(sv - mnew);
      l[r]  = fmaf(l[r],  al, p);
      a0[r] = fmaf(a0[r], al, p * Vv.x);
      a1[r] = fmaf(a1[r], al, p * Vv.y);
      a2[r] = fmaf(a2[r], al, p * Vv.z);
      m[r]  = mnew;
    }
  }

  // Epilogue: merge the 16 lane-local states per row (once).
#pragma unroll
  for (int r = 0; r < 8; ++r) {
    float mrow = m[r];
#pragma unroll
    for (int off = 1; off < 16; off <<= 1)
      mrow = fmaxf(mrow, __shfl_xor(mrow, off, 32));
    float f  = __builtin_amdgcn_exp2f(m[r] - mrow);
    float lr = l[r] * f;
    float b0 = a0[r] * f;
    float b1 = a1[r] * f;
    float b2 = a2[r] * f;
#pragma unroll
    for (int off = 1; off < 16; off <<= 1) {
      lr += __shfl_xor(lr, off, 32);
      b0 += __shfl_xor(b0, off, 32);
      b1 += __shfl_xor(b1, off, 32);
      b2 += __shfl_xor(b2, off, 32);
    }
    if (col == r) {
      int row = s0 + r + 8 * half;
      size_t ob = ((size_t)b * kS + row) * 9 + 3 * n;
      float inv = 1.0f / lr;
      O[ob + 0] = b0 * inv;
      O[ob + 1] = b1 * inv;
      O[ob + 2] = b2 * inv;
    }
  }
}

// ---------------------------------------------------------------------------
// Kernel 3a: per-block partial sums / sums-of-squares of the (49152, 9) matrix
// (deterministic two-stage reduction; no float atomics).
// ---------------------------------------------------------------------------
__global__ __launch_bounds__(256) void mbfca_stats1(const float* __restrict__ O,
                                                    float* __restrict__ part)
{
  __shared__ float red[256];
  int tid = threadIdx.x;
  float ls[9], lq[9];
#pragma unroll
  for (int j = 0; j < 9; ++j) { ls[j] = 0.f; lq[j] = 0.f; }
  int row0 = blockIdx.x * 1024;
  for (int it = 0; it < 4; ++it) {
    const float* o = O + (size_t)(row0 + it * 256 + tid) * 9;
#pragma unroll
    for (int j = 0; j < 9; ++j) { float v = o[j]; ls[j] += v; lq[j] += v * v; }
  }
#pragma unroll
  for (int j = 0; j < 18; ++j) {
    red[tid] = (j < 9) ? ls[j] : lq[j - 9];
    __syncthreads();
    for (int s = 128; s > 0; s >>= 1) {
      if (tid < s) red[tid] += red[tid + s];
      __syncthreads();
    }
    if (tid == 0) part[blockIdx.x * 18 + j] = red[0];
    __syncthreads();
  }
}

// Kernel 3b: finish reduction -> mean[9], inv_std[9]
__global__ void mbfca_stats2(const float* __restrict__ part, float* __restrict__ stats)
{
  int j = threadIdx.x;
  if (j < 9) {
    float s = 0.f, q = 0.f;
    for (int blk = 0; blk < 48; ++blk) {
      s += part[blk * 18 + j];
      q += part[blk * 18 + 9 + j];
    }
    float mean = s / (float)kRows;
    float var = q / (float)kRows - mean * mean;   // biased var, matches jnp.var
    stats[j] = mean;
    stats[9 + j] = rsqrtf(var + 1e-5f);
  }
}

// ---------------------------------------------------------------------------
// Kernel 4: BatchNorm + 9->3 FC per row.
// ---------------------------------------------------------------------------
__global__ __launch_bounds__(256) void mbfca_fc(
    const float* __restrict__ O, const float* __restrict__ stats,
    const float* __restrict__ gamma, const float* __restrict__ beta,
    const float* __restrict__ fcw, const float* __restrict__ fcb,
    float* __restrict__ out)
{
  int row = blockIdx.x * blockDim.x + threadIdx.x;
  if (row >= kRows) return;
  const float* o = O + (size_t)row * 9;
  float h[9];
#pragma unroll
  for (int j = 0; j < 9; ++j)
    h[j] = (o[j] - stats[j]) * stats[9 + j] * gamma[j] + beta[j];
#pragma unroll
  for (int i = 0; i < 3; ++i) {
    float acc = fcb[i];
#pragma unroll
    for (int j = 0; j < 9; ++j) acc = fmaf(h[j], fcw[i * 9 + j], acc);
    out[row * 3 + i] = acc;
  }
}

extern "C" void kernel_launch(void* const* d_in, const int* in_sizes, int n_in,
                              void* d_out, int out_size, void* d_ws, size_t ws_size,
                              hipStream_t stream)
{
  const float* x   = (const float*)d_in[0];
  const float* WQ  = (const float*)d_in[1];
  const float* bQ  = (const float*)d_in[2];
  const float* WK  = (const float*)d_in[3];
  const float* bK  = (const float*)d_in[4];
  const float* WV  = (const float*)d_in[5];
  const float* bV  = (const float*)d_in[6];
  const float* gam = (const float*)d_in[7];
  const float* bet = (const float*)d_in[8];
  const float* fcw = (const float*)d_in[9];
  const float* fcb = (const float*)d_in[10];
  float* out = (float*)d_out;

  float* ws = (float*)d_ws;
  float* Qp = ws;                                  // 147456*4
  float* Kp = Qp + (size_t)kNPos * 4;              // 147456*4
  float* Vp = Kp + (size_t)kNPos * 4;              // 147456*4
  float* O  = Vp + (size_t)kNPos * 4;              // 49152*9
  float* part  = O + (size_t)kRows * 9;            // 48*18
  float* stats = part + 48 * 18;                   // 18

  mbfca_proj<<<(kNPos + 255) / 256, 256, 0, stream>>>(x, WQ, bQ, WK, bK, WV, bV, Qp, Kp, Vp);
  mbfca_attn<<<(3 * kB * (kS / 16)) / 8, 256, 0, stream>>>(Qp, Kp, Vp, O);
  mbfca_stats1<<<48, 256, 0, stream>>>(O, part);
  mbfca_stats2<<<1, 32, 0, stream>>>(part, stats);
  mbfca_fc<<<(kRows + 255) / 256, 256, 0, stream>>>(O, stats, gam, bet, fcw, fcb, out);
}